// SDgraphConv_9964324127378
// MI455X (gfx1250) — compile-verified
//
#include <hip/hip_runtime.h>
#include <hip/hip_bf16.h>
#include <cstddef>

#define B_   32
#define NPG  1024
#define NN   (B_ * NPG)      // 32768 nodes
#define D_   128
#define DE_  16
#define DEG_ 16
#define EE   (NN * DEG_)

typedef float v2f __attribute__((ext_vector_type(2)));
typedef float v8f __attribute__((ext_vector_type(8)));

__device__ __forceinline__ v8f wmma4(v2f a, v2f b, v8f c) {
  // D = A(16x4,f32) * B(4x16,f32) + C(16x16,f32)
  return __builtin_amdgcn_wmma_f32_16x16x4_f32(false, a, false, b, (short)0, c, false, false);
}

__device__ __forceinline__ float lrelu(float v) { return v >= 0.0f ? v : 0.2f * v; }
__device__ __forceinline__ float ninf() { return -__builtin_inff(); }

// ---- guarded async global->LDS copy (gfx1250 GLOBAL_LOAD_ASYNC_TO_LDS path) ----
#ifdef __has_builtin
#  if __has_builtin(__builtin_amdgcn_load_to_lds)
#    define G2L_MODE 1
#  elif __has_builtin(__builtin_amdgcn_global_load_lds)
#    define G2L_MODE 2
#  else
#    define G2L_MODE 0
#  endif
#else
#  define G2L_MODE 0
#endif

__device__ __forceinline__ void cp_g2l(float* lp, const float* gp) {
#if G2L_MODE == 1
  __builtin_amdgcn_load_to_lds((void*)gp, (__attribute__((address_space(3))) void*)lp, 4, 0, 0);
#elif G2L_MODE == 2
  __builtin_amdgcn_global_load_lds((const __attribute__((address_space(1))) unsigned int*)gp,
                                   (__attribute__((address_space(3))) unsigned int*)lp, 4, 0, 0);
#else
  *lp = *gp;
#endif
}
__device__ __forceinline__ void g2l_fence() {
#if G2L_MODE
  asm volatile("s_wait_asynccnt 0" ::: "memory");
#endif
}

// ---------------------------------------------------------------------------
// Kernel 1: static conv message + segment_max (dst = e/16, contiguous).
// One block (256 thr = 8 waves) per 2 nodes: A = [x[src[e]], edge_attr[e]]
// 32x144 in LDS; each B fragment feeds two WMMAs (rows 0-15 / 16-31).
// ---------------------------------------------------------------------------
#define KMSG 144
#define KMSGP 148
__global__ void k_msg_agg(const float* __restrict__ x, const int* __restrict__ src,
                          const float* __restrict__ ea, const float* __restrict__ Wm,
                          const float* __restrict__ bm, float* __restrict__ agg) {
  __shared__ float As[32 * KMSGP];
  const int node0 = blockIdx.x * 2;
  const int e0 = node0 * DEG_;
  __builtin_prefetch((const char*)Wm + (size_t)threadIdx.x * 128, 0, 0);
  for (int f = threadIdx.x; f < 32 * KMSG; f += 256) {
    int r = f / KMSG, c = f % KMSG;
    float* lp = &As[r * KMSGP + c];
    if (c < D_) cp_g2l(lp, &x[(size_t)src[e0 + r] * D_ + c]);
    else        cp_g2l(lp, &ea[(size_t)(e0 + r) * DE_ + (c - D_)]);
  }
  g2l_fence();
  __syncthreads();
  const int wv = threadIdx.x >> 5, lane = threadIdx.x & 31;
  const int hl = lane >> 4, lm = lane & 15;
  const int col = wv * 16 + lm;
  v8f acc0 = {}, acc1 = {};
  for (int k = 0; k < KMSG; k += 4) {
    v2f a0, a1, b;
    b.x = Wm[(size_t)(k + 2 * hl) * D_ + col];
    b.y = Wm[(size_t)(k + 2 * hl + 1) * D_ + col];
    a0.x = As[lm * KMSGP + k + 2 * hl];
    a0.y = As[lm * KMSGP + k + 2 * hl + 1];
    a1.x = As[(16 + lm) * KMSGP + k + 2 * hl];
    a1.y = As[(16 + lm) * KMSGP + k + 2 * hl + 1];
    acc0 = wmma4(a0, b, acc0);
    acc1 = wmma4(a1, b, acc1);
  }
  const float bb = bm[col];
  float m0 = ninf(), m1 = ninf();
#pragma unroll
  for (int r = 0; r < 8; ++r) {
    m0 = fmaxf(m0, lrelu(acc0[r] + bb));
    m1 = fmaxf(m1, lrelu(acc1[r] + bb));
  }
  m0 = fmaxf(m0, __shfl_xor(m0, 16));         // combine row-halves
  m1 = fmaxf(m1, __shfl_xor(m1, 16));
  if (hl == 0) {
    agg[(size_t)node0 * D_ + col]       = m0;
    agg[(size_t)(node0 + 1) * D_ + col] = m1;
  }
}

// ---------------------------------------------------------------------------
// Kernel 2: generic concat-GEMM  out = act(cat @ W + b). Rows tiled by 16,
// cols by 128 (8 waves x 16). A row tile staged (async) in dynamic LDS.
// ---------------------------------------------------------------------------
struct Cat {
  const float* p[5];
  int w[5];
  int n;
};

__global__ void k_cat_gemm(Cat cat, const float* __restrict__ W,
                           const float* __restrict__ bias, float* __restrict__ out,
                           int K, int ncols, int use_leaky) {
  extern __shared__ float As[];
  const int KP = K + 4;
  const int row0 = blockIdx.x * 16;
  __builtin_prefetch((const char*)W + (size_t)threadIdx.x * 128, 0, 0);
  for (int f = threadIdx.x; f < 16 * K; f += 256) {
    int r = f / K, c = f % K;
    int s = 0, cc = c;
    while (cc >= cat.w[s]) { cc -= cat.w[s]; ++s; }
    cp_g2l(&As[r * KP + c], &cat.p[s][(size_t)(row0 + r) * cat.w[s] + cc]);
  }
  g2l_fence();
  __syncthreads();
  const int wv = threadIdx.x >> 5, lane = threadIdx.x & 31;
  const int hl = lane >> 4, lm = lane & 15;
  const int col = blockIdx.y * 128 + wv * 16 + lm;
  v8f acc = {};
  for (int k = 0; k < K; k += 4) {
    v2f a, b;
    a.x = As[lm * KP + k + 2 * hl];
    a.y = As[lm * KP + k + 2 * hl + 1];
    b.x = W[(size_t)(k + 2 * hl) * ncols + col];
    b.y = W[(size_t)(k + 2 * hl + 1) * ncols + col];
    acc = wmma4(a, b, acc);
  }
  const float bb = bias[col];
#pragma unroll
  for (int r = 0; r < 8; ++r) {
    float v = acc[r] + bb;
    if (use_leaky) v = lrelu(v);
    out[(size_t)(row0 + r + 8 * hl) * ncols + col] = v;
  }
}

// ---------------------------------------------------------------------------
// Kernel 3: row L2-normalize (xn = x / (||x|| + 1e-8)), one 128-thr block/row
// ---------------------------------------------------------------------------
__global__ void k_norm(const float* __restrict__ xin, float* __restrict__ xn) {
  __shared__ float red[128];
  const int row = blockIdx.x;
  float v = xin[(size_t)row * D_ + threadIdx.x];
  red[threadIdx.x] = v * v;
  __syncthreads();
  for (int s = 64; s > 0; s >>= 1) {
    if ((int)threadIdx.x < s) red[threadIdx.x] += red[threadIdx.x + s];
    __syncthreads();
  }
  const float nrm = sqrtf(red[0]) + 1e-8f;
  xn[(size_t)row * D_ + threadIdx.x] = v / nrm;
}

// ---------------------------------------------------------------------------
// Kernel 4: per-group cosine-sim (WMMA 16x16 tiles over K=128) + exact top-16.
// Block handles 16 rows of one group; 8 waves sweep the 64 column tiles,
// keeping per-wave top-16 lists in LDS; final merge by 16 threads.
// ---------------------------------------------------------------------------
__global__ void k_topk(const float* __restrict__ xn, int* __restrict__ idx_out) {
  __shared__ float As[16 * 132];
  __shared__ float tileS[8][256];
  __shared__ float lv[8][16][16];
  __shared__ int   li[8][16][16];
  const int row0 = blockIdx.x * 16;
  const int gbase = (row0 / NPG) * NPG;
  const int wv = threadIdx.x >> 5, lane = threadIdx.x & 31;
  const int hl = lane >> 4, lm = lane & 15;

  for (int f = threadIdx.x; f < 16 * D_; f += 256) {
    int r = f >> 7, c = f & 127;
    cp_g2l(&As[r * 132 + c], &xn[(size_t)(row0 + r) * D_ + c]);
  }
  for (int q = lane; q < 256; q += 32) {
    lv[wv][q >> 4][q & 15] = ninf();
    li[wv][q >> 4][q & 15] = gbase;
  }
  g2l_fence();
  __syncthreads();

  float curmin = ninf();
  for (int t = 0; t < 8; ++t) {                 // 8 tiles per wave, uniform loop
    const int ct = t * 8 + wv;
    const int c0 = ct * 16;
    v8f acc = {};
    const float* bp = &xn[(size_t)(gbase + c0 + lm) * D_];
    for (int k = 0; k < D_; k += 4) {
      v2f a, b;
      a.x = As[lm * 132 + k + 2 * hl];
      a.y = As[lm * 132 + k + 2 * hl + 1];
      b.x = bp[k + 2 * hl];
      b.y = bp[k + 2 * hl + 1];
      acc = wmma4(a, b, acc);
    }
    const int colnode = gbase + c0 + lm;
#pragma unroll
    for (int r = 0; r < 8; ++r) {
      const int rownode = row0 + r + 8 * hl;
      tileS[wv][(r + 8 * hl) * 16 + lm] = (rownode == colnode) ? ninf() : acc[r];
    }
    __syncthreads();
    if (hl == 0) {                              // lanes 0..15: one row each
      const int row = lm;
      for (int j = 0; j < 16; ++j) {
        const float v = tileS[wv][row * 16 + j];
        if (v > curmin) {
          int am = 0; float mv = lv[wv][row][0];
          for (int q = 1; q < 16; ++q) { float qv = lv[wv][row][q]; if (qv < mv) { mv = qv; am = q; } }
          lv[wv][row][am] = v;
          li[wv][row][am] = gbase + c0 + j;
          mv = lv[wv][row][0];
          for (int q = 1; q < 16; ++q) mv = fminf(mv, lv[wv][row][q]);
          curmin = mv;
        }
      }
    }
    __syncthreads();
  }

  if (threadIdx.x < 16) {                       // merge 8 wave lists -> top16
    const int row = threadIdx.x;
    float fv[16]; int fi[16];
    for (int q = 0; q < 16; ++q) { fv[q] = ninf(); fi[q] = gbase; }
    float cm = ninf();
    for (int w8 = 0; w8 < 8; ++w8)
      for (int q = 0; q < 16; ++q) {
        const float v = lv[w8][row][q];
        if (v > cm) {
          int am = 0; float mv = fv[0];
          for (int z = 1; z < 16; ++z) if (fv[z] < mv) { mv = fv[z]; am = z; }
          fv[am] = v; fi[am] = li[w8][row][q];
          mv = fv[0];
          for (int z = 1; z < 16; ++z) mv = fminf(mv, fv[z]);
          cm = mv;
        }
      }
    const int node = row0 + row;
    for (int q = 0; q < 16; ++q) idx_out[(size_t)node * 16 + q] = fi[q];
  }
}

// ---------------------------------------------------------------------------
// Kernel 5: dynamic conv message + max over k=16 neighbors. One block per
// 2 nodes: A = [xi, xj - xi] 32x256, K=256; each B fragment feeds two WMMAs.
// ---------------------------------------------------------------------------
__global__ void k_dyn_msg(const float* __restrict__ xb, const int* __restrict__ idx,
                          const float* __restrict__ W, const float* __restrict__ bvec,
                          float* __restrict__ dout) {
  __shared__ float As[32 * 260];
  __shared__ int nb[32];
  const int node0 = blockIdx.x * 2;
  __builtin_prefetch((const char*)W + (size_t)threadIdx.x * 128, 0, 0);
  if (threadIdx.x < 32) nb[threadIdx.x] = idx[(size_t)node0 * 16 + threadIdx.x];
  __syncthreads();
  for (int f = threadIdx.x; f < 32 * 256; f += 256) {
    int r = f >> 8, c = f & 255;
    const int nd = node0 + (r >> 4);
    const float xi = xb[(size_t)nd * D_ + (c & 127)];
    float v;
    if (c < D_) v = xi;
    else        v = xb[(size_t)nb[r] * D_ + (c - D_)] - xi;
    As[r * 260 + c] = v;
  }
  __syncthreads();
  const int wv = threadIdx.x >> 5, lane = threadIdx.x & 31;
  const int hl = lane >> 4, lm = lane & 15;
  const int col = wv * 16 + lm;
  v8f acc0 = {}, acc1 = {};
  for (int k = 0; k < 256; k += 4) {
    v2f a0, a1, b;
    b.x = W[(size_t)(k + 2 * hl) * D_ + col];
    b.y = W[(size_t)(k + 2 * hl + 1) * D_ + col];
    a0.x = As[lm * 260 + k + 2 * hl];
    a0.y = As[lm * 260 + k + 2 * hl + 1];
    a1.x = As[(16 + lm) * 260 + k + 2 * hl];
    a1.y = As[(16 + lm) * 260 + k + 2 * hl + 1];
    acc0 = wmma4(a0, b, acc0);
    acc1 = wmma4(a1, b, acc1);
  }
  const float bb = bvec[col];
  float m0 = ninf(), m1 = ninf();
#pragma unroll
  for (int r = 0; r < 8; ++r) {
    m0 = fmaxf(m0, lrelu(acc0[r] + bb));
    m1 = fmaxf(m1, lrelu(acc1[r] + bb));
  }
  m0 = fmaxf(m0, __shfl_xor(m0, 16));
  m1 = fmaxf(m1, __shfl_xor(m1, 16));
  if (hl == 0) {
    dout[(size_t)node0 * D_ + col]       = m0;
    dout[(size_t)(node0 + 1) * D_ + col] = m1;
  }
}

// ---------------------------------------------------------------------------
extern "C" void kernel_launch(void* const* d_in, const int* in_sizes, int n_in,
                              void* d_out, int out_size, void* d_ws, size_t ws_size,
                              hipStream_t stream) {
  const float* x    = (const float*)d_in[0];
  const int*   src  = (const int*)d_in[1];        // edge_index row 0
  const float* ea   = (const float*)d_in[2];
  const float* Wm1  = (const float*)d_in[4];
  const float* bm1  = (const float*)d_in[5];
  const float* Wu1  = (const float*)d_in[6];
  const float* bu1  = (const float*)d_in[7];
  const float* Wm2  = (const float*)d_in[8];
  const float* bm2  = (const float*)d_in[9];
  const float* Wu2  = (const float*)d_in[10];
  const float* bu2  = (const float*)d_in[11];
  const float* dgW1 = (const float*)d_in[12];
  const float* dgb1 = (const float*)d_in[13];
  const float* dgW2 = (const float*)d_in[14];
  const float* dgb2 = (const float*)d_in[15];
  const float* fW1  = (const float*)d_in[16];
  const float* fb1  = (const float*)d_in[17];
  const float* fW2  = (const float*)d_in[18];
  const float* fb2  = (const float*)d_in[19];

  const size_t NB = (size_t)NN * D_;              // 4.19M floats per feature map
  float* agg = (float*)d_ws;
  float* sg1 = agg + NB;
  float* sg2 = sg1 + NB;
  float* d1  = sg2 + NB;
  float* d2  = d1 + NB;
  float* xn  = d2 + NB;
  float* f1  = xn + NB;                           // N x 256
  int*   idx = (int*)(f1 + 2 * NB);

  const dim3 blk(256);
  const dim3 gRow(NN / 16, 1);

  // static conv 1
  k_msg_agg<<<NN / 2, blk, 0, stream>>>(x, src, ea, Wm1, bm1, agg);
  {
    Cat c; c.p[0] = x;   c.p[1] = agg; c.w[0] = 128; c.w[1] = 128; c.n = 2;
    k_cat_gemm<<<gRow, blk, 16 * 260 * sizeof(float), stream>>>(c, Wu1, bu1, sg1, 256, 128, 1);
  }
  // static conv 2
  k_msg_agg<<<NN / 2, blk, 0, stream>>>(sg1, src, ea, Wm2, bm2, agg);
  {
    Cat c; c.p[0] = sg1; c.p[1] = agg; c.w[0] = 128; c.w[1] = 128; c.n = 2;
    k_cat_gemm<<<gRow, blk, 16 * 260 * sizeof(float), stream>>>(c, Wu2, bu2, sg2, 256, 128, 1);
  }
  // dynamic conv 1 (on sg1)
  k_norm<<<NN, 128, 0, stream>>>(sg1, xn);
  k_topk<<<NN / 16, blk, 0, stream>>>(xn, idx);
  k_dyn_msg<<<NN / 2, blk, 0, stream>>>(sg1, idx, dgW1, dgb1, d1);
  // dynamic conv 2 (on d1)
  k_norm<<<NN, 128, 0, stream>>>(d1, xn);
  k_topk<<<NN / 16, blk, 0, stream>>>(xn, idx);
  k_dyn_msg<<<NN / 2, blk, 0, stream>>>(d1, idx, dgW2, dgb2, d2);
  // fusion MLP
  {
    Cat c; c.p[0] = sg1; c.p[1] = d1; c.p[2] = d2; c.p[3] = sg2;
    c.w[0] = c.w[1] = c.w[2] = c.w[3] = 128; c.n = 4;
    k_cat_gemm<<<dim3(NN / 16, 2), blk, 16 * 516 * sizeof(float), stream>>>(c, fW1, fb1, f1, 512, 256, 1);
  }
  {
    Cat c; c.p[0] = f1; c.p[1] = sg1; c.p[2] = d1; c.p[3] = d2; c.p[4] = sg2;
    c.w[0] = 256; c.w[1] = c.w[2] = c.w[3] = c.w[4] = 128; c.n = 5;
    k_cat_gemm<<<gRow, blk, 16 * 772 * sizeof(float), stream>>>(c, fW2, fb2, (float*)d_out, 768, 128, 0);
  }
}